// hyperTem_60756607369791
// MI455X (gfx1250) — compile-verified
//
#include <hip/hip_runtime.h>

typedef float v2f __attribute__((ext_vector_type(2)));
typedef float v8f __attribute__((ext_vector_type(8)));

#define B_ 32
#define T_ 24
#define N_ 2048
#define D_ 64
#define K_ 32
#define H_ 16

// Workspace layout (floats):
//   adjw  [H][T][N]                  offset 0        size 786432   (3 MB)
//   Wf    [B*T][64 frag][32 lane][2] offset 786432   size 3145728  (12.6 MB, WMMA-B fragment layout)
//   biasw [B*T][64]                  offset 3932160  size 49152
#define WF_OFF   786432
#define BIAS_OFF 3932160

// ---------------- Kernel 1: adj_dyn[h,t,n] = sum_k node_emb[n,k] * adj[k,h,t]
__global__ __launch_bounds__(256) void k_adj(const float* __restrict__ node_emb,
                                             const float* __restrict__ adj,
                                             float* __restrict__ adjw) {
    const int g = blockIdx.x * 256 + threadIdx.x;    // g < H*T*N = 786432
    const int n = g % N_;
    const int t = (g / N_) % T_;
    const int h = g / (N_ * T_);
    float acc = 0.f;
#pragma unroll
    for (int k = 0; k < K_; ++k)
        acc += node_emb[n * K_ + k] * adj[(k * H_ + h) * T_ + t];
    adjw[g] = acc;
}

// ---------------- Kernel 2: W[b,t,i,o] = sum_k time_eb[b,t,k]*wpool[k,i,o]
// stored directly in WMMA-B fragment order: e = ((ki*4+oj)*32 + lane)*2 + p
//   with i = 4*ki + 2*(lane>>4) + p,  o = 16*oj + (lane&15)
__global__ __launch_bounds__(256) void k_wgen(const float* __restrict__ time_eb,
                                              const float* __restrict__ wpool,
                                              const float* __restrict__ bpool,
                                              float* __restrict__ Wf,
                                              float* __restrict__ biasw) {
    const int bt = blockIdx.x;                       // b*T + t
    __shared__ float teb[K_];
    if (threadIdx.x < K_) teb[threadIdx.x] = time_eb[bt * K_ + threadIdx.x];
    __syncthreads();

    for (int j = 0; j < 16; ++j) {
        const int e = threadIdx.x + 256 * j;         // < 4096, consecutive tid -> coalesced store
        const int p = e & 1;
        const int L = (e >> 1) & 31;
        const int c = e >> 6;                        // ki*4 + oj, 0..63
        const int i = 4 * (c >> 2) + 2 * (L >> 4) + p;
        const int o = 16 * (c & 3) + (L & 15);
        float acc = 0.f;
#pragma unroll
        for (int k = 0; k < K_; ++k)
            acc += teb[k] * wpool[(k * D_ + i) * D_ + o];
        Wf[(size_t)bt * 4096 + e] = acc;
    }
    if (threadIdx.x < D_) {
        const int o = threadIdx.x;
        float acc = 0.f;
#pragma unroll
        for (int k = 0; k < K_; ++k)
            acc += teb[k] * bpool[k * D_ + o];
        biasw[bt * D_ + o] = acc;
    }
}

// ---------------- Kernel 3: fused  M_n = A^T A,  ret = M_n eb,  out = leaky(ret@W + bias + eb)
// LDS (floats): Ms [16 n][24 t][24 t2]            @ 0      (9216)
//               ret[24 t][16 n][stride 68]        @ 9216   (26112)  -- As tile overlapped here
#define MS_OFF 0
#define RET_OFF 9216
#define AS_OFF 9216
#define RET_STRIDE 68
#define SMEM_FLOATS 35328   // 141312 bytes

__global__ __launch_bounds__(256) void k_fused(const float* __restrict__ eb,
                                               const float* __restrict__ adjw,
                                               const float* __restrict__ Wf,
                                               const float* __restrict__ biasw,
                                               float* __restrict__ out) {
    extern __shared__ float smem[];
    const int b   = blockIdx.x >> 7;                 // N/16 = 128 tiles per batch
    const int n0  = (blockIdx.x & 127) << 4;
    const int tid = threadIdx.x;

    // ---- Phase 1: stage A tile: As[h][t][n] (16x24x16)
    for (int j = 0; j < 24; ++j) {
        const int idx = tid + 256 * j;               // < 6144
        const int n = idx & 15;
        const int t = (idx >> 4) % 24;
        const int h = idx / 384;
        smem[AS_OFF + (h * T_ + t) * 16 + n] = adjw[(size_t)(h * T_ + t) * N_ + n0 + n];
    }
    __syncthreads();

    // ---- Phase 2: Ms[n][t][t2] = sum_h As[h][t][n] * As[h][t2][n]
    for (int j = 0; j < 36; ++j) {
        const int e  = tid + 256 * j;                // < 9216
        const int n  = e / 576;
        const int r  = e % 576;
        const int t  = r / 24;
        const int t2 = r % 24;
        float acc = 0.f;
#pragma unroll
        for (int h = 0; h < H_; ++h)
            acc += smem[AS_OFF + (h * T_ + t) * 16 + n] * smem[AS_OFF + (h * T_ + t2) * 16 + n];
        smem[MS_OFF + e] = acc;                      // Ms region disjoint from As region
    }
    __syncthreads();

    // ---- Phase 3: ret[t][n][d] = sum_t2 Ms[n][t][t2] * eb[b][t2][n0+n][d]
    {
        const int n  = tid >> 4;                     // 0..15
        const int dg = (tid & 15) << 2;              // d-group of 4, coalesced float4 loads
        float rets[T_][4];
#pragma unroll
        for (int t = 0; t < T_; ++t)
#pragma unroll
            for (int q = 0; q < 4; ++q) rets[t][q] = 0.f;

        for (int t2 = 0; t2 < T_; ++t2) {
            const float4 e4 = *(const float4*)(eb + (((size_t)(b * T_ + t2) * N_ + n0 + n) * D_ + dg));
#pragma unroll
            for (int t = 0; t < T_; ++t) {
                const float m = smem[MS_OFF + (n * T_ + t) * T_ + t2];
                rets[t][0] += m * e4.x;
                rets[t][1] += m * e4.y;
                rets[t][2] += m * e4.z;
                rets[t][3] += m * e4.w;
            }
        }
        // spill ret tile to LDS (overlaps dead As region); 16B-aligned b128 stores
#pragma unroll
        for (int t = 0; t < T_; ++t) {
            float4 v = make_float4(rets[t][0], rets[t][1], rets[t][2], rets[t][3]);
            *(float4*)(&smem[RET_OFF + (t * 16 + n) * RET_STRIDE + dg]) = v;
        }
    }
    __syncthreads();

    // ---- Phase 4: out[t][n][o] = leaky( sum_i ret[t][n][i]*W[b,t,i,o] + bias + eb )
    const int w = tid >> 5;                          // wave 0..7
    const int L = tid & 31;                          // lane
    for (int tw = 0; tw < 3; ++tw) {
        const int t = w + 8 * tw;                    // 8 waves x 3 = 24 t-values
        v8f acc[4];
#pragma unroll
        for (int oj = 0; oj < 4; ++oj)
#pragma unroll
            for (int r = 0; r < 8; ++r) acc[oj][r] = 0.f;

        const float* wbase = Wf + (size_t)(b * T_ + t) * 4096;
#pragma unroll
        for (int ki = 0; ki < 16; ++ki) {
            // A fragment (16x4 fp32): lane L -> row L&15, K = 2*(L>>4) + {0,1}
            const v2f a = *(const v2f*)(&smem[RET_OFF + (t * 16 + (L & 15)) * RET_STRIDE
                                             + 4 * ki + 2 * (L >> 4)]);
#pragma unroll
            for (int oj = 0; oj < 4; ++oj) {
                const v2f bf = *(const v2f*)(wbase + (((ki * 4 + oj) * 32 + L) << 1));
                acc[oj] = __builtin_amdgcn_wmma_f32_16x16x4_f32(
                    false, a, false, bf, (short)0, acc[oj], false, false);
            }
        }
        // epilogue: C/D layout: VGPR r -> row r + 8*(L>>4), col = L&15
        const int nrow0 = n0 + 8 * (L >> 4);
#pragma unroll
        for (int oj = 0; oj < 4; ++oj) {
            const int o = oj * 16 + (L & 15);
            const float bv = biasw[(b * T_ + t) * D_ + o];
#pragma unroll
            for (int r = 0; r < 8; ++r) {
                const size_t gi = ((size_t)(b * T_ + t) * N_ + nrow0 + r) * D_ + o;
                const float x = acc[oj][r] + bv + eb[gi];
                out[gi] = (x > 0.f) ? x : 0.01f * x;
            }
        }
    }
}

extern "C" void kernel_launch(void* const* d_in, const int* in_sizes, int n_in,
                              void* d_out, int out_size, void* d_ws, size_t ws_size,
                              hipStream_t stream) {
    const float* eb       = (const float*)d_in[0];
    const float* node_emb = (const float*)d_in[1];
    const float* time_eb  = (const float*)d_in[2];
    const float* adj      = (const float*)d_in[3];
    const float* wpool    = (const float*)d_in[4];
    const float* bpool    = (const float*)d_in[5];
    float* ws    = (float*)d_ws;
    float* adjw  = ws;
    float* Wf    = ws + WF_OFF;
    float* biasw = ws + BIAS_OFF;
    float* outp  = (float*)d_out;

    k_adj<<<(H_ * T_ * N_) / 256, 256, 0, stream>>>(node_emb, adj, adjw);
    k_wgen<<<B_ * T_, 256, 0, stream>>>(time_eb, wpool, bpool, Wf, biasw);
    k_fused<<<B_ * (N_ / 16), 256, SMEM_FLOATS * 4, stream>>>(eb, adjw, Wf, biasw, outp);
}